// BIGMem_15023795602046
// MI455X (gfx1250) — compile-verified
//
#include <hip/hip_runtime.h>
#include <math.h>

// ---------------- problem constants ----------------
#define Hdim   768
#define KTOP   10
#define RNUM   12
#define READK  100
#define MDim   64
#define WKn    100
#define INTERn 3072
#define Nn_TOT 50000
#define Bn     4
#define En     128
#define BEn    512        // B*E
#define SLOTSn 110        // KTOP + READK
#define CANDn  1000       // KTOP*WKn
#define KBch   64         // A-chunk depth staged in LDS per step

typedef float v2f __attribute__((ext_vector_type(2)));
typedef float v8f __attribute__((ext_vector_type(8)));

__device__ __forceinline__ float gelu_exact(float x) {
  return 0.5f * x * (1.0f + erff(x * 0.70710678118654752f));
}

// ---------------- zero e22n + loss ----------------
__global__ void k_zero(float* __restrict__ e22n, size_t n, float* __restrict__ loss) {
  size_t i = (size_t)blockIdx.x * blockDim.x + threadIdx.x;
  size_t stride = (size_t)gridDim.x * blockDim.x;
  for (; i < n; i += stride) e22n[i] = 0.0f;
  if (blockIdx.x == 0 && threadIdx.x == 0) *loss = 0.0f;
}

// ---------------- f32 WMMA GEMM: C[M,N] = act(A[M,K] @ W[K,N] + bias) ----------------
// Block = 128 threads (4 waves) computing a 16x256 strip (wave w -> cols [w*64,w*64+64)).
// A chunk (16 x 64 f32 = 4KB) is staged in LDS with gfx1250 async-LDS DMA
// (global_load_async_to_lds_b128), double buffered so the DMA of chunk c+1 overlaps
// the 64 WMMAs of chunk c. All 4 waves share the LDS A tile (4x less A HBM traffic).
// B fragments are software-pipelined one k-step ahead so WMMAs overlap the B loads.
// Fragment layouts (V_WMMA_F32_16X16X4_F32):
//   A 16x4: lanes0-15 row M=l (v0=K0,v1=K1); lanes16-31 same rows (v0=K2,v1=K3).
//   B 4x16: lanes0-15 col N=l (v0=K0,v1=K1); lanes16-31 same cols (v0=K2,v1=K3).
//   C/D 16x16: lanes0-15 N=l, VGPR v -> M=v; lanes16-31 -> M=8+v.
__global__ void k_gemm_wmma(const float* __restrict__ A, const float* __restrict__ W,
                            const float* __restrict__ bias, float* __restrict__ C,
                            int M, int Kd, int Nw, int act) {
  const int tid  = threadIdx.x;     // 128
  const int lane = tid & 31;
  const int wave = tid >> 5;
  const int nb   = Nw >> 8;         // N-blocks of 256 cols
  const int mt   = blockIdx.x / nb;
  const int ntb  = blockIdx.x - mt * nb;
  const int l15  = lane & 15;
  const int lh   = lane >> 4;
  const int col0 = ntb * 256 + wave * 64 + l15;

  __shared__ __align__(16) float asub[2][16 * KBch];
  const unsigned ldsbase = (unsigned)(size_t)&asub[0][0];  // LDS aperture: offset in addr[31:0]

  // each thread copies 2 x 16B segments of the 4KB chunk (256 segs, 128 threads)
  const int seg0 = tid, seg1 = tid + 128;
  const size_t arowbytes = (size_t)Kd * 4;
  const char* gA = (const char*)(A + (size_t)(mt * 16) * Kd);
  const char* g0base = gA + (size_t)(seg0 >> 4) * arowbytes + (size_t)(seg0 & 15) * 16;
  const char* g1base = gA + (size_t)(seg1 >> 4) * arowbytes + (size_t)(seg1 & 15) * 16;
  const unsigned l0base = ldsbase + (unsigned)seg0 * 16;
  const unsigned l1base = ldsbase + (unsigned)seg1 * 16;

#define ISSUE_ACOPY(bufi, kb)                                                       \
  do {                                                                              \
    const char* g0 = g0base + (size_t)(kb) * 4;                                     \
    const char* g1 = g1base + (size_t)(kb) * 4;                                     \
    unsigned lo0 = l0base + (unsigned)(bufi) * (16 * KBch * 4);                     \
    unsigned lo1 = l1base + (unsigned)(bufi) * (16 * KBch * 4);                     \
    asm volatile("global_load_async_to_lds_b128 %0, %1, off" ::"v"(lo0), "v"(g0)    \
                 : "memory");                                                       \
    asm volatile("global_load_async_to_lds_b128 %0, %1, off" ::"v"(lo1), "v"(g1)    \
                 : "memory");                                                       \
  } while (0)

  v8f acc[4] = {};
  const int nch = Kd / KBch;
  const size_t wstep = (size_t)4 * Nw;  // floats per k-step (4 rows of W)

  ISSUE_ACOPY(0, 0);
  for (int c = 0; c < nch; ++c) {
    const int kb = c * KBch;
    if (c + 1 < nch) {
      ISSUE_ACOPY((c + 1) & 1, kb + KBch);
      // prefetch next chunk of W for this wave's columns
      __builtin_prefetch(W + (size_t)(kb + KBch + lh * 2) * Nw + col0, 0, 0);
      asm volatile("s_wait_asynccnt 0x2" ::: "memory");  // drain chunk c only
    } else {
      asm volatile("s_wait_asynccnt 0x0" ::: "memory");
    }
    __syncthreads();  // chunk c visible to all waves

    const float* as = &asub[c & 1][0] + l15 * KBch + lh * 2;
    const float* wp = W + (size_t)(kb + lh * 2) * Nw + col0;

    // preload B fragments for k-step 0 of this chunk
    v2f b0, b1, b2, b3;
    b0.x = wp[0];  b0.y = wp[Nw];
    b1.x = wp[16]; b1.y = wp[16 + Nw];
    b2.x = wp[32]; b2.y = wp[32 + Nw];
    b3.x = wp[48]; b3.y = wp[48 + Nw];

#pragma unroll 4
    for (int k0 = 0; k0 < KBch; k0 += 4) {
      v2f a = *(const v2f*)(as + k0);  // ds_load_b64
      v2f n0 = b0, n1 = b1, n2 = b2, n3 = b3;
      if (k0 + 4 < KBch) {             // issue next step's B loads before WMMAs
        const float* wn = wp + wstep;
        n0.x = wn[0];  n0.y = wn[Nw];
        n1.x = wn[16]; n1.y = wn[16 + Nw];
        n2.x = wn[32]; n2.y = wn[32 + Nw];
        n3.x = wn[48]; n3.y = wn[48 + Nw];
        wp = wn;
      }
      acc[0] = __builtin_amdgcn_wmma_f32_16x16x4_f32(false, a, false, b0, (short)0, acc[0], false, false);
      acc[1] = __builtin_amdgcn_wmma_f32_16x16x4_f32(false, a, false, b1, (short)0, acc[1], false, false);
      acc[2] = __builtin_amdgcn_wmma_f32_16x16x4_f32(false, a, false, b2, (short)0, acc[2], false, false);
      acc[3] = __builtin_amdgcn_wmma_f32_16x16x4_f32(false, a, false, b3, (short)0, acc[3], false, false);
      b0 = n0; b1 = n1; b2 = n2; b3 = n3;
    }
    __syncthreads();  // all waves done reading buf[c&1] before it is re-filled
  }
#undef ISSUE_ACOPY

#pragma unroll
  for (int j = 0; j < 4; ++j) {
    const int col = ntb * 256 + wave * 64 + j * 16 + l15;
    const float bc = bias ? bias[col] : 0.0f;
#pragma unroll
    for (int v = 0; v < 8; ++v) {
      const int row = mt * 16 + lh * 8 + v;
      float x = acc[j][v] + bc;
      if (act) x = gelu_exact(x);
      C[(size_t)row * Nw + col] = x;
    }
  }
}

// ---------------- hierarchical product-key indexer ----------------
// one 128-thread block per (b,e) row; q_idx layout [BE, 4*64]
__global__ void k_indexer(const float* __restrict__ qidx,
                          const float* __restrict__ key0, const float* __restrict__ key1,
                          const float* __restrict__ key2, const float* __restrict__ key3,
                          float* __restrict__ e2n_val, int* __restrict__ e2n_ind) {
  const int row = blockIdx.x;
  const int tid = threadIdx.x;  // 128
  __shared__ float q[256];
  __shared__ float sc[128];
  __shared__ float av[128];
  __shared__ float val[KTOP];
  __shared__ int   ind[KTOP];

  for (int i = tid; i < 256; i += 128) q[i] = qidx[(size_t)row * 256 + i];
  __syncthreads();

  // level 0: 100 keys
  if (tid < 100) {
    const float* kp = key0 + (size_t)tid * 64;
    float s = 0.0f;
    for (int d = 0; d < 64; ++d) s += q[d] * kp[d];
    sc[tid] = s * 0.125f;  // 1/sqrt(64)
  }
  __syncthreads();
  if (tid == 0) {
    for (int kk = 0; kk < KTOP; ++kk) {      // top-10 (ties -> lowest index)
      float best = -3.0e38f; int bi = 0;
      for (int n2 = 0; n2 < 100; ++n2)
        if (sc[n2] > best) { best = sc[n2]; bi = n2; }
      val[kk] = best; ind[kk] = bi; sc[bi] = -3.0e38f;
    }
    // softmax-over-100 + renorm-over-10 == softmax over the top-10 scores
    float m = val[0], Z = 0.0f;
    for (int kk = 0; kk < KTOP; ++kk) { float e = expf(val[kk] - m); val[kk] = e; Z += e; }
    float iZ = 1.0f / Z;
    for (int kk = 0; kk < KTOP; ++kk) val[kk] *= iZ;
  }
  __syncthreads();

  for (int lvl = 1; lvl < 4; ++lvl) {
    const int branch = (lvl == 3) ? 5 : 10;
    const int nc = KTOP * branch;
    const float* qv = q + lvl * 64;
    if (tid < nc) {
      const int k = tid / branch, c = tid - k * branch;
      const int node = ind[k];
      const float* kp;
      if (lvl == 1)      kp = key1 + ((size_t)node * 10 + c) * 64;
      else if (lvl == 2) kp = key2 + ((size_t)node * 10 + c) * 64;
      else               kp = key3 + ((size_t)node * 5 + c) * 64;
      float s = 0.0f;
      for (int d = 0; d < 64; ++d) s += qv[d] * kp[d];
      sc[tid] = s * 0.125f;
    }
    __syncthreads();
    if (tid == 0) {
      for (int k = 0; k < KTOP; ++k) {       // per-node softmax over children
        float m = -3.0e38f;
        for (int c = 0; c < branch; ++c) m = fmaxf(m, sc[k * branch + c]);
        float Z = 0.0f;
        for (int c = 0; c < branch; ++c) { float e = expf(sc[k * branch + c] - m); av[k * branch + c] = e; Z += e; }
        float sfac = val[k] / Z;
        for (int c = 0; c < branch; ++c) av[k * branch + c] *= sfac;
      }
      float nval[KTOP]; int nf[KTOP]; int nid[KTOP];
      for (int kk = 0; kk < KTOP; ++kk) {
        float best = -3.0e38f; int bi = 0;
        for (int t = 0; t < nc; ++t)
          if (av[t] > best) { best = av[t]; bi = t; }
        nval[kk] = best; nf[kk] = bi; av[bi] = -3.0e38f;
      }
      float Z = 0.0f;
      for (int kk = 0; kk < KTOP; ++kk) Z += nval[kk];
      float iZ = 1.0f / Z;
      for (int kk = 0; kk < KTOP; ++kk) {
        int f = nf[kk];
        nid[kk] = ind[f / branch] * branch + (f - (f / branch) * branch);
      }
      for (int kk = 0; kk < KTOP; ++kk) { ind[kk] = nid[kk]; val[kk] = nval[kk] * iZ; }
    }
    __syncthreads();
  }
  if (tid < KTOP) {
    e2n_val[(size_t)row * KTOP + tid] = val[tid];
    e2n_ind[(size_t)row * KTOP + tid] = ind[tid];
  }
}

// ---------------- scatter-add candidates into dense e22n[row, N] ----------------
__global__ void k_scatter(const float* __restrict__ e2n_val, const int* __restrict__ e2n_ind,
                          const int* __restrict__ we_r, const float* __restrict__ ww_r,
                          float* __restrict__ e22n) {
  const int t = blockIdx.x * 256 + threadIdx.x;
  if (t >= BEn * CANDn) return;
  const int row = t / CANDn;
  const int c = t - row * CANDn;
  const int k = c / WKn, w = c - k * WKn;
  const int node = e2n_ind[row * KTOP + k];
  const int col = we_r[(size_t)node * WKn + w];
  const float v = e2n_val[row * KTOP + k] * ww_r[(size_t)node * WKn + w];
  atomicAdd(&e22n[(size_t)row * Nn_TOT + col], v);
}

// ---------------- e222e + write-loss partials (one thread per (b,i,j)) ----------------
__global__ void k_loss(const float* __restrict__ e22n, const float* __restrict__ e2n_val,
                       const int* __restrict__ e2n_ind, const float* __restrict__ e2e,
                       int r, float* __restrict__ part) {
  const int t = blockIdx.x * 256 + threadIdx.x;  // < 65536 = B*E*E
  const int b = t >> 14;
  const int ij = t & 16383;
  const int i = ij >> 7, j = ij & 127;
  const int rowi = b * En + i, rowj = b * En + j;
  const float* er = e22n + (size_t)rowi * Nn_TOT;
  float s = 0.0f;
#pragma unroll
  for (int k = 0; k < KTOP; ++k)
    s += er[e2n_ind[rowj * KTOP + k]] * e2n_val[rowj * KTOP + k];
  const float d = s - e2e[(((size_t)b * RNUM + r) * En + i) * En + j];
  float sq = d * d * (1.0f / 65536.0f);
  __shared__ float red[256];
  red[threadIdx.x] = sq; __syncthreads();
  for (int off = 128; off > 0; off >>= 1) {
    if (threadIdx.x < off) red[threadIdx.x] += red[threadIdx.x + off];
    __syncthreads();
  }
  if (threadIdx.x == 0) part[blockIdx.x] = red[0];
}

__global__ void k_loss_final(const float* __restrict__ part, float* __restrict__ loss) {
  __shared__ float red[256];
  red[threadIdx.x] = part[threadIdx.x]; __syncthreads();
  for (int off = 128; off > 0; off >>= 1) {
    if (threadIdx.x < off) red[threadIdx.x] += red[threadIdx.x + off];
    __syncthreads();
  }
  if (threadIdx.x == 0) *loss += red[0];
}

// ---------------- drain e22n (atomicExch->0 dedups+zeros) + top-100 selection ----------------
__device__ __forceinline__ bool cand_better(float v1, int c1, int s1,
                                            float v2, int c2, int s2) {
  if (v1 != v2) return v1 > v2;
  if (c1 != c2) return c1 < c2;
  return s1 < s2;
}

__global__ void k_collect_topk(float* __restrict__ e22n, const float* __restrict__ e2n_val,
                               const int* __restrict__ e2n_ind, const int* __restrict__ we_r,
                               int* __restrict__ all_ind, int r) {
  const int row = blockIdx.x, tid = threadIdx.x;  // 256 threads
  __shared__ float vals[CANDn];
  __shared__ int   cols[CANDn];
  __shared__ float bvv[256];
  __shared__ int   bcc[256];
  __shared__ int   bss[256];

  for (int c = tid; c < CANDn; c += 256) {
    const int k = c / WKn, w = c - k * WKn;
    const int node = e2n_ind[row * KTOP + k];
    const int col = we_r[(size_t)node * WKn + w];
    vals[c] = atomicExch(&e22n[(size_t)row * Nn_TOT + col], 0.0f);  // value once, dups get 0; zeros buffer
    cols[c] = col;
  }
  __syncthreads();

  int* outp = all_ind + ((size_t)row * RNUM + r) * SLOTSn;
  if (tid < KTOP) outp[tid] = e2n_ind[row * KTOP + tid];

  for (int it = 0; it < READK; ++it) {
    float mv = vals[tid]; int mc = cols[tid]; int ms = tid;
    for (int c = tid + 256; c < CANDn; c += 256)
      if (cand_better(vals[c], cols[c], c, mv, mc, ms)) { mv = vals[c]; mc = cols[c]; ms = c; }
    bvv[tid] = mv; bcc[tid] = mc; bss[tid] = ms;
    __syncthreads();
    for (int off = 128; off > 0; off >>= 1) {
      if (tid < off &&
          cand_better(bvv[tid + off], bcc[tid + off], bss[tid + off],
                      bvv[tid], bcc[tid], bss[tid])) {
        bvv[tid] = bvv[tid + off]; bcc[tid] = bcc[tid + off]; bss[tid] = bss[tid + off];
      }
      __syncthreads();
    }
    if (tid == 0) { outp[KTOP + it] = bcc[0]; vals[bss[0]] = -3.0e38f; }
    __syncthreads();
  }
}

// ---------------- attention over 110 gathered value slots (per row,r) ----------------
__global__ void k_attn(const float* __restrict__ qh, const int* __restrict__ all_ind,
                       const float* __restrict__ memory_values, float* __restrict__ attd) {
  const int row = blockIdx.x / RNUM;
  const int r = blockIdx.x - row * RNUM;
  const int tid = threadIdx.x;  // 128
  __shared__ float q[64];
  __shared__ float sc[SLOTSn];
  __shared__ float w[SLOTSn];
  const int* idx = all_ind + ((size_t)row * RNUM + r) * SLOTSn;
  const float* mv = memory_values + (size_t)r * Nn_TOT * MDim;
  if (tid < 64) q[tid] = qh[(size_t)row * Hdim + r * MDim + tid];
  __syncthreads();
  if (tid < SLOTSn) {
    const float* vp = mv + (size_t)idx[tid] * MDim;
    float s = 0.0f;
    for (int d = 0; d < MDim; ++d) s += q[d] * vp[d];
    sc[tid] = s * 0.125f;  // 1/sqrt(64)
  }
  __syncthreads();
  if (tid == 0) {
    float m = -3.0e38f;
    for (int s = 0; s < SLOTSn; ++s) m = fmaxf(m, sc[s]);
    float Z = 0.0f;
    for (int s = 0; s < SLOTSn; ++s) { float e = expf(sc[s] - m); w[s] = e; Z += e; }
    float iZ = 1.0f / Z;
    for (int s = 0; s < SLOTSn; ++s) w[s] *= iZ;
  }
  __syncthreads();
  if (tid < 64) {
    float acc = 0.0f;
    for (int s = 0; s < SLOTSn; ++s) acc += w[s] * mv[(size_t)idx[s] * MDim + tid];
    attd[(size_t)row * Hdim + r * MDim + tid] = acc;
  }
}

// ---------------- layernorm over H=768 per row ----------------
__global__ void k_ln(const float* __restrict__ h, const float* __restrict__ g,
                     const float* __restrict__ b, float* __restrict__ out) {
  const int row = blockIdx.x, tid = threadIdx.x;  // 256
  __shared__ float red[256];
  __shared__ float stat[2];
  const float* hr = h + (size_t)row * Hdim;
  float s = 0.0f;
  for (int i = tid; i < Hdim; i += 256) s += hr[i];
  red[tid] = s; __syncthreads();
  for (int off = 128; off > 0; off >>= 1) {
    if (tid < off) red[tid] += red[tid + off];
    __syncthreads();
  }
  if (tid == 0) stat[0] = red[0] * (1.0f / Hdim);
  __syncthreads();
  const float mu = stat[0];
  float v = 0.0f;
  for (int i = tid; i < Hdim; i += 256) { float d = hr[i] - mu; v += d * d; }
  red[tid] = v; __syncthreads();
  for (int off = 128; off > 0; off >>= 1) {
    if (tid < off) red[tid] += red[tid + off];
    __syncthreads();
  }
  if (tid == 0) stat[1] = rsqrtf(red[0] * (1.0f / Hdim) + 1e-5f);
  __syncthreads();
  const float rstd = stat[1];
  for (int i = tid; i < Hdim; i += 256)
    out[(size_t)row * Hdim + i] = (hr[i] - mu) * rstd * g[i] + b[i];
}

// ---------------- launch ----------------
extern "C" void kernel_launch(void* const* d_in, const int* in_sizes, int n_in,
                              void* d_out, int out_size, void* d_ws, size_t ws_size,
                              hipStream_t stream) {
  (void)in_sizes; (void)n_in; (void)ws_size;
  const float* elem_hiddens   = (const float*)d_in[0];
  const float* elems_to_elems = (const float*)d_in[1];
  const float* Wq_idx = (const float*)d_in[2];
  const float* bq_idx = (const float*)d_in[3];
  const float* key0 = (const float*)d_in[4];
  const float* key1 = (const float*)d_in[5];
  const float* key2 = (const float*)d_in[6];
  const float* key3 = (const float*)d_in[7];
  // d_in[8] = memory_keys (gathered but unused by the reference output)
  const float* memory_values  = (const float*)d_in[9];
  const float* wiring_weights = (const float*)d_in[10];
  const float* Wq_mha = (const float*)d_in[11];
  const float* bq_mha = (const float*)d_in[12];
  const float* W1 = (const float*)d_in[13];
  const float* b1 = (const float*)d_in[14];
  const float* W2 = (const float*)d_in[15];
  const float* b2 = (const float*)d_in[16];
  const float* ln_g = (const float*)d_in[17];
  const float* ln_b = (const float*)d_in[18];
  const int* wiring_edges = (const int*)d_in[19];

  float* out  = (float*)d_out;
  float* loss = out + (out_size - 1);

  // workspace carve-out (~117 MB)
  float* ws = (float*)d_ws;
  float* q_idx     = ws;            ws += (size_t)BEn * 256;
  float* qh        = ws;            ws += (size_t)BEn * Hdim;
  float* e2n_val   = ws;            ws += (size_t)BEn * KTOP;
  int*   e2n_ind   = (int*)ws;      ws += (size_t)BEn * KTOP;
  int*   all_ind   = (int*)ws;      ws += (size_t)BEn * RNUM * SLOTSn;
  float* attd      = ws;            ws += (size_t)BEn * Hdim;
  float* h1        = ws;            ws += (size_t)BEn * INTERn;
  float* h2        = ws;            ws += (size_t)BEn * Hdim;
  float* loss_part = ws;            ws += 256;
  float* e22n      = ws;            // BEn * Nn_TOT floats (102.4 MB, fits L2)

  k_zero<<<4096, 256, 0, stream>>>(e22n, (size_t)BEn * Nn_TOT, loss);

  // q_idx = x @ Wq_idx + bq_idx     [512,768]x[768,256]
  k_gemm_wmma<<<(BEn / 16) * (256 / 256), 128, 0, stream>>>(
      elem_hiddens, Wq_idx, bq_idx, q_idx, BEn, Hdim, 256, 0);

  k_indexer<<<BEn, 128, 0, stream>>>(q_idx, key0, key1, key2, key3, e2n_val, e2n_ind);

  // qh = x @ Wq_mha + bq_mha        [512,768]x[768,768]
  k_gemm_wmma<<<(BEn / 16) * (Hdim / 256), 128, 0, stream>>>(
      elem_hiddens, Wq_mha, bq_mha, qh, BEn, Hdim, Hdim, 0);

  for (int r = 0; r < RNUM; ++r) {
    const int* we_r = wiring_edges + (size_t)r * Nn_TOT * WKn;
    const float* ww_r = wiring_weights + (size_t)r * Nn_TOT * WKn;
    k_scatter<<<(BEn * CANDn + 255) / 256, 256, 0, stream>>>(e2n_val, e2n_ind, we_r, ww_r, e22n);
    k_loss<<<256, 256, 0, stream>>>(e22n, e2n_val, e2n_ind, elems_to_elems, r, loss_part);
    k_loss_final<<<1, 256, 0, stream>>>(loss_part, loss);
    k_collect_topk<<<BEn, 256, 0, stream>>>(e22n, e2n_val, e2n_ind, we_r, all_ind, r);
  }

  k_attn<<<BEn * RNUM, 128, 0, stream>>>(qh, all_ind, memory_values, attd);

  // h1 = gelu(attd @ W1 + b1)       [512,768]x[768,3072]
  k_gemm_wmma<<<(BEn / 16) * (INTERn / 256), 128, 0, stream>>>(
      attd, W1, b1, h1, BEn, Hdim, INTERn, 1);
  // h2 = h1 @ W2 + b2               [512,3072]x[3072,768]
  k_gemm_wmma<<<(BEn / 16) * (Hdim / 256), 128, 0, stream>>>(
      h1, W2, b2, h2, BEn, INTERn, Hdim, 0);

  k_ln<<<BEn, 256, 0, stream>>>(h2, ln_g, ln_b, out);
}